// DGLHGNNConv_30142080484149
// MI455X (gfx1250) — compile-verified
//
#include <hip/hip_runtime.h>

typedef __attribute__((ext_vector_type(2))) float v2f;
typedef __attribute__((ext_vector_type(8))) float v8f;

#define IN_CH 128
#define HO    128

// ---------------------------------------------------------------------------
// Zero-fill (grid-stride) — d_out / Xe are poisoned by the harness.
// ---------------------------------------------------------------------------
__global__ void zero_f32(float* __restrict__ p, int n) {
  int i = blockIdx.x * blockDim.x + threadIdx.x;
  int stride = gridDim.x * blockDim.x;
  for (; i < n; i += stride) p[i] = 0.0f;
}

// ---------------------------------------------------------------------------
// Xp = X @ Wlin using V_WMMA_F32_16X16X4_F32 (full fp32 precision).
// One wave computes one 16x16 output tile; K-loop steps by 4 (32 WMMAs/tile).
//
// A (16x4 f32, 2 VGPRs): lane&15 = row M, lane>>4 picks K pair {0,1} vs {2,3}.
// B (4x16 f32, 2 VGPRs): mirrored (lane&15 = col N, lane>>4 picks K pair).
// C/D (16x16 f32, 8 VGPRs): vgpr r -> row m0 + (lane>>4)*8 + r, col lane&15.
// ---------------------------------------------------------------------------
__global__ __launch_bounds__(256) void gemm_wmma_f32(
    const float* __restrict__ X, const float* __restrict__ Wl,
    float* __restrict__ Xp, int M) {
  const int ntiles = HO / 16;  // 8
  int wave = (blockIdx.x * blockDim.x + threadIdx.x) >> 5;
  int lane = threadIdx.x & 31;
  int mt = wave / ntiles;
  int nt = wave - mt * ntiles;
  int m0 = mt * 16;
  int n0 = nt * 16;
  if (m0 >= M) return;  // wave-uniform

  int half = lane >> 4;   // 0 or 1
  int l15  = lane & 15;
  int arow = m0 + l15;
  int bcol = n0 + l15;

  if (m0 + 16 <= M) {
    // Full-tile WMMA path (EXEC all ones — required for WMMA).
    v8f acc = {};
    for (int k0 = 0; k0 < IN_CH; k0 += 4) {
      int ka = k0 + half * 2;
      v2f a, b;
      a.x = X[(size_t)arow * IN_CH + ka];
      a.y = X[(size_t)arow * IN_CH + ka + 1];
      b.x = Wl[(size_t)ka * HO + bcol];
      b.y = Wl[(size_t)(ka + 1) * HO + bcol];
      acc = __builtin_amdgcn_wmma_f32_16x16x4_f32(
          /*neg_a=*/false, a, /*neg_b=*/false, b,
          /*c_mod=*/(short)0, acc, /*reuse_a=*/false, /*reuse_b=*/false);
    }
#pragma unroll
    for (int r = 0; r < 8; ++r) {
      Xp[(size_t)(m0 + half * 8 + r) * HO + bcol] = acc[r];
    }
  } else {
    // Scalar tail (not hit for M=100000, kept for robustness).
    if (half == 0 && arow < M) {
      for (int n = 0; n < 16; ++n) {
        float s = 0.0f;
        for (int k = 0; k < IN_CH; ++k)
          s += X[(size_t)arow * IN_CH + k] * Wl[(size_t)k * HO + (n0 + n)];
        Xp[(size_t)arow * HO + (n0 + n)] = s;
      }
    }
  }
}

// ---------------------------------------------------------------------------
// Stage 1: Xe[dst] += Xp[src]   (copy_u_sum, node -> hyperedge)
// One wave per edge, 4 channels per lane (float4 coalesced gather,
// global_atomic_add_f32 scatter into L2-resident Xe).
// ---------------------------------------------------------------------------
__global__ __launch_bounds__(128) void scatter_v2e(
    const float* __restrict__ Xp, const int* __restrict__ src,
    const int* __restrict__ dst, float* __restrict__ Xe, int nnz) {
  int t = blockIdx.x * blockDim.x + threadIdx.x;
  int e = t >> 5;
  if (e >= nnz) return;
  int c = (t & 31) * 4;
  int s = src[e];
  int d = dst[e];
  const float4 v = *reinterpret_cast<const float4*>(Xp + (size_t)s * HO + c);
  float* o = Xe + (size_t)d * HO + c;
  atomicAdd(o + 0, v.x);
  atomicAdd(o + 1, v.y);
  atomicAdd(o + 2, v.z);
  atomicAdd(o + 3, v.w);
}

// ---------------------------------------------------------------------------
// Stage 2 fused: out[src] += degV[src] * degE[dst] * W[dst] * Xe[dst]
// (degE/W/degV scaling folded into the per-edge contribution — identical
//  algebra since the scales are constant per source/destination row.)
// ---------------------------------------------------------------------------
__global__ __launch_bounds__(128) void scatter_e2v(
    const float* __restrict__ Xe, const float* __restrict__ degE,
    const float* __restrict__ W, const float* __restrict__ degV,
    const int* __restrict__ src, const int* __restrict__ dst,
    float* __restrict__ out, int nnz) {
  int t = blockIdx.x * blockDim.x + threadIdx.x;
  int e = t >> 5;
  if (e >= nnz) return;
  int c = (t & 31) * 4;
  int s = src[e];
  int d = dst[e];
  float scale = degE[d] * W[d] * degV[s];
  const float4 v = *reinterpret_cast<const float4*>(Xe + (size_t)d * HO + c);
  float* o = out + (size_t)s * HO + c;
  atomicAdd(o + 0, v.x * scale);
  atomicAdd(o + 1, v.y * scale);
  atomicAdd(o + 2, v.z * scale);
  atomicAdd(o + 3, v.w * scale);
}

// ---------------------------------------------------------------------------
extern "C" void kernel_launch(void* const* d_in, const int* in_sizes, int n_in,
                              void* d_out, int out_size, void* d_ws, size_t ws_size,
                              hipStream_t stream) {
  const float* X    = (const float*)d_in[0];   // [N, 128]
  const float* Wl   = (const float*)d_in[1];   // [128, 128]
  const float* degE = (const float*)d_in[2];   // [E, 1]
  const float* degV = (const float*)d_in[3];   // [N, 1]
  const float* W    = (const float*)d_in[4];   // [E, 1]
  const int* g1_src = (const int*)d_in[5];     // [nnz]
  const int* g1_dst = (const int*)d_in[6];     // [nnz]
  float* out = (float*)d_out;

  const int N   = in_sizes[0] / IN_CH;  // 100000
  const int E   = in_sizes[2];          // 25000
  const int nnz = in_sizes[5];          // 1600000

  float* Xp = (float*)d_ws;             // N * 128 floats
  float* Xe = Xp + (size_t)N * HO;      // E * 128 floats

  // Zero accumulation buffers (harness poisons them; must re-zero each call).
  zero_f32<<<2048, 256, 0, stream>>>(Xe, E * HO);
  zero_f32<<<4096, 256, 0, stream>>>(out, N * HO);

  // GEMM: one wave per 16x16 tile.
  {
    int mtiles = (N + 15) / 16;
    int waves = mtiles * (HO / 16);
    int blocks = (waves * 32 + 255) / 256;
    gemm_wmma_f32<<<blocks, 256, 0, stream>>>(X, Wl, Xp, N);
  }

  // Scatter stages: one wave (32 lanes x float4) per edge.
  {
    int blocks = (nnz + 3) / 4;  // 4 edges per 128-thread block
    scatter_v2e<<<blocks, 128, 0, stream>>>(Xp, g1_src, g1_dst, Xe, nnz);
    scatter_e2v<<<blocks, 128, 0, stream>>>(Xe, degE, W, degV,
                                            g1_src, g1_dst, out, nnz);
  }
}